// Generator_18038862643863
// MI455X (gfx1250) — compile-verified
//
#include <hip/hip_runtime.h>

typedef __attribute__((ext_vector_type(2))) float v2f;
typedef __attribute__((ext_vector_type(8))) float v8f;

// ---------------------------------------------------------------------------
// Stage 1: h = concat(noise, labels) @ W_lin^T + b_lin
//   M=64 (4 tiles), N=8192 (512 tiles), K=101 padded to 104 (26 wmma steps).
//   One wave per 16x16 output tile; fp32 WMMA keeps exact reference math.
// ---------------------------------------------------------------------------
__global__ __launch_bounds__(256) void gen_linear_wmma(
    const float* __restrict__ noise,    // (64,100)
    const float* __restrict__ labels,   // (64,)
    const float* __restrict__ Wlin,     // (8192,101) row-major
    const float* __restrict__ blin,     // (8192,)
    float* __restrict__ h)              // (64,8192)
{
  const int wave   = (blockIdx.x * blockDim.x + threadIdx.x) >> 5;
  const int lane   = threadIdx.x & 31;
  const int tile_n = wave & 511;   // 512 N tiles
  const int tile_m = wave >> 9;    // 4 M tiles
  const int half   = lane >> 4;
  const int mn     = lane & 15;
  const int m      = tile_m * 16 + mn;   // A row for this lane
  const int n      = tile_n * 16 + mn;   // B/C column for this lane

  v8f acc = {0.f, 0.f, 0.f, 0.f, 0.f, 0.f, 0.f, 0.f};
  for (int kk = 0; kk < 26; ++kk) {
    v2f a, b;
#pragma unroll
    for (int j = 0; j < 2; ++j) {
      const int k = kk * 4 + half * 2 + j;
      float av = 0.f, bv = 0.f;
      if (k < 100)       av = noise[m * 100 + k];
      else if (k == 100) av = labels[m];
      if (k < 101)       bv = Wlin[n * 101 + k];
      a[j] = av;
      b[j] = bv;
    }
    acc = __builtin_amdgcn_wmma_f32_16x16x4_f32(false, a, false, b,
                                                (short)0, acc, false, false);
  }

  const float bias = blin[n];
#pragma unroll
  for (int r = 0; r < 8; ++r) {
    const int mm = tile_m * 16 + half * 8 + r;  // C/D row = r + 8*half
    h[mm * 8192 + n] = acc[r] + bias;
  }
}

// ---------------------------------------------------------------------------
// Stage 2/3: 4D transposed conv (stride 1, pad 1, k=3) as implicit GEMM.
//   y[n,o,d] = b[o] + sum_{c,tap} W[c,o,flip(tap)] * x[n,c,d+tap-1]
//   M = 64*8*8*8*16 = 524288 positions (32768 tiles of 16)
//   N = Cout (2 or 4) padded into a 16-wide tile
//   K = Cin*81, padded to a multiple of 4
//   Kernel flip: flipping all four base-3 digits of tap t is just (80 - t),
//   so B loads index the torch (Cin,Cout,3,3,3,3) weight directly.
// ---------------------------------------------------------------------------
template <int CIN, int COUT>
__global__ __launch_bounds__(256) void convT4d_wmma(
    const float* __restrict__ x,    // (64, CIN, 8,8,8,16)
    const float* __restrict__ w,    // (CIN, COUT, 3,3,3,3)
    const float* __restrict__ bias, // (COUT,)
    float* __restrict__ y)          // (64, COUT, 8,8,8,16)
{
  constexpr int KDIM  = CIN * 81;
  constexpr int KITER = (KDIM + 3) / 4;

  const int wave = (blockIdx.x * blockDim.x + threadIdx.x) >> 5;
  const int lane = threadIdx.x & 31;
  const int half = lane >> 4;
  const int mn   = lane & 15;

  // A-row (output position) decomposition, fixed per lane for the whole loop
  const int m  = wave * 16 + mn;        // 0..524287
  const int nb = m >> 13;               // batch index
  const int s  = m & 8191;              // spatial index within (8,8,8,16)
  const int d1 = s >> 10;
  const int d2 = (s >> 7) & 7;
  const int d3 = (s >> 4) & 7;
  const int d4 = s & 15;
  const int o  = mn;                    // output channel (column); real if < COUT

  v8f acc = {0.f, 0.f, 0.f, 0.f, 0.f, 0.f, 0.f, 0.f};
  for (int kk = 0; kk < KITER; ++kk) {
    v2f a, b;
#pragma unroll
    for (int j = 0; j < 2; ++j) {
      const int k = kk * 4 + half * 2 + j;
      float av = 0.f, bv = 0.f;
      if (k < KDIM) {
        const int c  = k / 81;
        const int t  = k - c * 81;      // tap index in [0,81)
        const int k1 = t / 27;
        const int rr = t - k1 * 27;
        const int k2 = rr / 9;
        const int k3 = (rr - k2 * 9) / 3;
        const int k4 = rr - k2 * 9 - k3 * 3;
        const int e1 = d1 + k1 - 1;
        const int e2 = d2 + k2 - 1;
        const int e3 = d3 + k3 - 1;
        const int e4 = d4 + k4 - 1;
        if (((unsigned)e1 < 8u) & ((unsigned)e2 < 8u) &
            ((unsigned)e3 < 8u) & ((unsigned)e4 < 16u)) {
          av = x[(nb * CIN + c) * 8192 + e1 * 1024 + e2 * 128 + e3 * 16 + e4];
        }
        if (o < COUT) {
          bv = w[(c * COUT + o) * 81 + (80 - t)];  // spatially flipped tap
        }
      }
      a[j] = av;
      b[j] = bv;
    }
    acc = __builtin_amdgcn_wmma_f32_16x16x4_f32(false, a, false, b,
                                                (short)0, acc, false, false);
  }

  if (o < COUT) {
    const float bo = bias[o];
#pragma unroll
    for (int r = 0; r < 8; ++r) {
      const int mm  = wave * 16 + half * 8 + r;  // output position
      const int nb2 = mm >> 13;
      const int s2  = mm & 8191;
      y[(nb2 * COUT + o) * 8192 + s2] = acc[r] + bo;
    }
  }
}

// ---------------------------------------------------------------------------
// Launch: linear -> conv1 -> conv2, intermediates staged in d_ws (6 MB << L2)
// ---------------------------------------------------------------------------
extern "C" void kernel_launch(void* const* d_in, const int* in_sizes, int n_in,
                              void* d_out, int out_size, void* d_ws, size_t ws_size,
                              hipStream_t stream) {
  (void)in_sizes; (void)n_in; (void)out_size; (void)ws_size;

  const float* noise  = (const float*)d_in[0];  // (64,100)
  const float* labels = (const float*)d_in[1];  // (64,)
  const float* Wlin   = (const float*)d_in[2];  // (8192,101)
  const float* blin   = (const float*)d_in[3];  // (8192,)
  const float* W1     = (const float*)d_in[4];  // (1,2,3,3,3,3)
  const float* b1     = (const float*)d_in[5];  // (2,)
  const float* W2     = (const float*)d_in[6];  // (2,4,3,3,3,3)
  const float* b2     = (const float*)d_in[7];  // (4,)

  float* h  = (float*)d_ws;        // 64*8192        = 2 MB
  float* y1 = h + 64 * 8192;       // 64*2*8192      = 4 MB
  float* y2 = (float*)d_out;       // 64*4*8192 output

  // Linear: 4*512 = 2048 waves -> 256 blocks x 8 waves
  gen_linear_wmma<<<256, 256, 0, stream>>>(noise, labels, Wlin, blin, h);
  // Convs: 524288/16 = 32768 waves -> 4096 blocks x 8 waves
  convT4d_wmma<1, 2><<<4096, 256, 0, stream>>>(h, W1, b1, y1);
  convT4d_wmma<2, 4><<<4096, 256, 0, stream>>>(y1, W2, b2, y2);
}